// HetSAGPooling_25151328485777
// MI455X (gfx1250) — compile-verified
//
#include <hip/hip_runtime.h>
#include <math.h>

typedef __attribute__((ext_vector_type(2))) float v2f;
typedef __attribute__((ext_vector_type(8))) float v8f;
typedef unsigned long long u64;
typedef unsigned int u32;

#define TPB 256

// ---------------------------------------------------------------- utilities

__device__ inline void atomicMaxF(float* addr, float val) {
  val = val + 0.0f;            // canonicalize -0.0 -> +0.0 (int-pattern trick breaks on -0.0)
  if (val >= 0.0f) {
    atomicMax((int*)addr, __float_as_int(val));
  } else {
    atomicMin((unsigned int*)addr, __float_as_uint(val));
  }
}

// ---------------------------------------------------------------- init

__global__ void init_kernel(float* amax, float* denom, float* outn, float* acc, int N) {
  int i = blockIdx.x * blockDim.x + threadIdx.x;
  if (i < N) {
    amax[i]  = -__builtin_inff();
    denom[i] = 0.0f;
    outn[i]  = 0.0f;
  }
  if (i < 8) acc[i] = 0.0f;
}

// ---------------------------------------------------------------- kqv via WMMA f32 16x16x4
// Per wave: 16 nodes (M) x 16 cols (N, cols 0..2 = k,q,v raw; 3..15 zero-padded),
// K-loop over F in steps of 4.
// A layout (16x4 f32): lanes 0-15 hold M=0..15 / K=kb+0,1 ; lanes 16-31 hold M=0..15 / K=kb+2,3.
// B layout (4x16 f32): same K split across lane halves, N = lane&15.
// C layout: VGPR r -> M=r (lanes 0-15) and M=r+8 (lanes 16-31), N = lane&15.
//
// W (F,3) is pre-swizzled into LDS in exactly the per-lane B-operand order:
//   sB[(kb/4)*64 + lane*2 + comp] = Wpad[kb + (lane>>4)*2 + comp][lane&15]
// so the inner loop is: 1x global b64 (A) + 1x ds b64 (B) + 1x v_wmma.
__global__ void kqv_wmma_kernel(const float* __restrict__ x,
                                const float* __restrict__ W,   // (F,3) row-major
                                const float* __restrict__ b,   // (3,)
                                const float* __restrict__ W_krel, // (2,1,1)
                                const float* __restrict__ b_krel, // (2,1)
                                const float* __restrict__ W_vrel,
                                const float* __restrict__ b_vrel,
                                int type, int cnt, int F, int goff,
                                float* __restrict__ qa,
                                float* __restrict__ ka,
                                float* __restrict__ va) {
  extern __shared__ float sB[];   // (F/4) * 64 floats = F*16 floats

  // cooperative preload + swizzle + zero-pad of W into LDS
  int total = F * 16;
  for (int idx = threadIdx.x; idx < total; idx += blockDim.x) {
    int kb4  = idx >> 6;
    int rem  = idx & 63;
    int l    = rem >> 1;
    int comp = rem & 1;
    int k    = kb4 * 4 + ((l >> 4) << 1) + comp;
    int col  = l & 15;
    sB[idx] = (col < 3) ? W[(size_t)k * 3 + col] : 0.0f;
  }
  __syncthreads();

  int wave = threadIdx.x >> 5;
  int lane = threadIdx.x & 31;
  int tile = blockIdx.x * (TPB / 32) + wave;
  int n    = lane & 15;          // A row (M) and B/C col (N)
  int half = lane >> 4;
  int row  = tile * 16 + n;
  int rowc = (row < cnt) ? row : (cnt - 1);   // clamp; EXEC stays all-1 for WMMA
  const float* xrow = x + (size_t)rowc * F;
  int koff = half * 2;

  v8f c = {};
#pragma unroll 4
  for (int kb = 0; kb < F; kb += 4) {
    v2f a  = *(const v2f*)(xrow + kb + koff);                 // global b64
    v2f bb = *(const v2f*)(sB + ((kb >> 2) << 6) + (lane << 1)); // ds b64, conflict-free
    c = __builtin_amdgcn_wmma_f32_16x16x4_f32(false, a, false, bb, (short)0, c, false, false);
  }

  float wk = W_krel[type], bk = b_krel[type];
  float wv = W_vrel[type], bv = b_vrel[type];
  int mAdd = half * 8;
  for (int r = 0; r < 8; ++r) {
    int m    = r + mAdd;
    int node = tile * 16 + m;
    if (n < 3 && node < cnt) {
      float val = c[r] + b[n];
      int g = goff + node;
      if (n == 0)      ka[g] = val * wk + bk;   // k = k_raw*W_krel + b_krel
      else if (n == 1) qa[g] = val;             // q
      else             va[g] = val * wv + bv;   // v = v_raw*W_vrel + b_vrel
    }
  }
}

// ---------------------------------------------------------------- edge passes

__device__ inline bool edge_decode(int e, int E1, int E2, int Nc,
                                   const int* __restrict__ s1, const int* __restrict__ d1,
                                   const int* __restrict__ s2, const int* __restrict__ d2,
                                   float p1, float p2,
                                   int& src, int& dst, float& attr) {
  if (e >= E1 + E2) return false;
  if (e < E1) { src = s1[e];          dst = Nc + d1[e];   attr = p1; }
  else        { int t = e - E1; src = Nc + s2[t]; dst = d2[t]; attr = p2; }
  return true;
}

__global__ void edge_max_kernel(const int* __restrict__ s1, const int* __restrict__ d1,
                                const int* __restrict__ s2, const int* __restrict__ d2,
                                const float* __restrict__ p1p, const float* __restrict__ p2p,
                                const float* __restrict__ qa, const float* __restrict__ ka,
                                float* __restrict__ amax, int E1, int E2, int Nc) {
  int e = blockIdx.x * blockDim.x + threadIdx.x;
  int src, dst; float attr;
  if (!edge_decode(e, E1, E2, Nc, s1, d1, s2, d2, p1p[0], p2p[0], src, dst, attr)) return;
  float alpha = qa[dst] * ka[src] * attr;     // D=1 -> /sqrt(D)=1
  atomicMaxF(&amax[dst], alpha);
}

__global__ void edge_denom_kernel(const int* __restrict__ s1, const int* __restrict__ d1,
                                  const int* __restrict__ s2, const int* __restrict__ d2,
                                  const float* __restrict__ p1p, const float* __restrict__ p2p,
                                  const float* __restrict__ qa, const float* __restrict__ ka,
                                  const float* __restrict__ amax,
                                  float* __restrict__ denom, int E1, int E2, int Nc) {
  int e = blockIdx.x * blockDim.x + threadIdx.x;
  int src, dst; float attr;
  if (!edge_decode(e, E1, E2, Nc, s1, d1, s2, d2, p1p[0], p2p[0], src, dst, attr)) return;
  float alpha = qa[dst] * ka[src] * attr;
  atomicAdd(&denom[dst], expf(alpha - amax[dst]));
}

__global__ void edge_out_kernel(const int* __restrict__ s1, const int* __restrict__ d1,
                                const int* __restrict__ s2, const int* __restrict__ d2,
                                const float* __restrict__ p1p, const float* __restrict__ p2p,
                                const float* __restrict__ qa, const float* __restrict__ ka,
                                const float* __restrict__ va,
                                const float* __restrict__ amax, const float* __restrict__ denom,
                                float* __restrict__ outn, int E1, int E2, int Nc) {
  int e = blockIdx.x * blockDim.x + threadIdx.x;
  int src, dst; float attr;
  if (!edge_decode(e, E1, E2, Nc, s1, d1, s2, d2, p1p[0], p2p[0], src, dst, attr)) return;
  float alpha = qa[dst] * ka[src] * attr;
  float w = expf(alpha - amax[dst]) / (denom[dst] + 1e-16f);
  atomicAdd(&outn[dst], va[src] * w);
}

// ---------------------------------------------------------------- out-proj + LN stats

__global__ void node_post_kernel(const float* __restrict__ outn,
                                 const float* __restrict__ Woc, const float* __restrict__ boc,
                                 const float* __restrict__ Won, const float* __restrict__ bon,
                                 float* __restrict__ y, float* __restrict__ acc,
                                 int N, int Nc) {
  __shared__ float r0[TPB], r1[TPB], r2[TPB], r3[TPB];
  int t = threadIdx.x;
  int i = blockIdx.x * TPB + t;
  float c0 = 0.f, c1 = 0.f, c2 = 0.f, c3 = 0.f;
  if (i < N) {
    int ty  = (i < Nc) ? 0 : 1;
    float w  = ty ? Won[0] : Woc[0];
    float bb = ty ? bon[0] : boc[0];
    float yy = outn[i] * w + bb;
    y[i] = yy;
    if (ty == 0) { c0 = yy; c1 = yy * yy; }
    else         { c2 = yy; c3 = yy * yy; }
  }
  r0[t] = c0; r1[t] = c1; r2[t] = c2; r3[t] = c3;
  __syncthreads();
  for (int s = TPB / 2; s > 0; s >>= 1) {
    if (t < s) { r0[t] += r0[t+s]; r1[t] += r1[t+s]; r2[t] += r2[t+s]; r3[t] += r3[t+s]; }
    __syncthreads();
  }
  if (t == 0) {
    atomicAdd(&acc[0], r0[0]); atomicAdd(&acc[1], r1[0]);
    atomicAdd(&acc[2], r2[0]); atomicAdd(&acc[3], r3[0]);
  }
}

// ---------------------------------------------------------------- LN + gelu + tanh + key build

__global__ void node_score_kernel(const float* __restrict__ y, const float* __restrict__ acc,
                                  const float* __restrict__ lnwc, const float* __restrict__ lnbc,
                                  const float* __restrict__ lnwn, const float* __restrict__ lnbn,
                                  const float* __restrict__ wsel,
                                  float* __restrict__ score, u64* __restrict__ keys,
                                  int N, int Nc, int Nn, int SORTN) {
  int i = blockIdx.x * blockDim.x + threadIdx.x;
  if (i >= SORTN) return;
  if (i >= N) { keys[i] = 0ull; return; }   // pad with minimal key
  int ty = (i < Nc) ? 0 : 1;
  float cntf = ty ? (float)Nn : (float)Nc;
  float m   = (ty ? acc[2] : acc[0]) / cntf;
  float sq  = (ty ? acc[3] : acc[1]) / cntf;
  float var = sq - m * m;
  float w  = ty ? lnwn[0] : lnwc[0];
  float bb = ty ? lnbn[0] : lnbc[0];
  float z  = w * (y[i] - m) * rsqrtf(var + 1e-5f) + bb;
  float a  = 0.5f * z * (1.0f + erff(z * 0.70710678118654752f));  // exact gelu
  float ws = wsel[0];
  float s  = tanhf(a * (ws * rsqrtf(ws * ws)));                   // * sign(w_sel)
  score[i] = s;
  u32 u  = __float_as_uint(s);
  u32 mu = (u & 0x80000000u) ? ~u : (u | 0x80000000u);  // monotone map: bigger uint = bigger float
  keys[i] = ((u64)mu << 32) | (u32)(~(u32)i);           // tie-break: smaller index wins
}

// ---------------------------------------------------------------- descending bitonic sort

__global__ void bitonic_kernel(u64* __restrict__ keys, int j, int kk) {
  int i = blockIdx.x * blockDim.x + threadIdx.x;
  int ixj = i ^ j;
  if (ixj > i) {
    u64 a = keys[i], b = keys[ixj];
    bool down = ((i & kk) == 0);              // descending segment
    if (down ? (a < b) : (a > b)) { keys[i] = b; keys[ixj] = a; }
  }
}

// ---------------------------------------------------------------- final gather * vals

__global__ void final_gather_kernel(const u64* __restrict__ keys,
                                    const float* __restrict__ score,
                                    const float* __restrict__ x_cell,
                                    const float* __restrict__ x_net,
                                    float* __restrict__ out,
                                    int Nc, int F) {
  int p = blockIdx.x;
  u64 key = keys[p];
  u32 node = ~(u32)(key & 0xFFFFFFFFull);
  float val = score[node];
  const float* xr = (node < (u32)Nc) ? (x_cell + (size_t)node * F)
                                     : (x_net + (size_t)(node - Nc) * F);
  for (int f = threadIdx.x; f < F; f += blockDim.x)
    out[(size_t)p * F + f] = xr[f] * val;
}

// ---------------------------------------------------------------- launch

extern "C" void kernel_launch(void* const* d_in, const int* in_sizes, int n_in,
                              void* d_out, int out_size, void* d_ws, size_t ws_size,
                              hipStream_t stream) {
  const float* x_cell = (const float*)d_in[0];
  const float* x_net  = (const float*)d_in[1];
  const int*   s1     = (const int*)d_in[2];
  const int*   d1     = (const int*)d_in[3];
  const int*   s2     = (const int*)d_in[4];
  const int*   d2     = (const int*)d_in[5];
  // d_in[6], d_in[7] = batch indices (single graph) -> unused
  const float* Wkqv_c = (const float*)d_in[8];
  const float* bkqv_c = (const float*)d_in[9];
  const float* Wkqv_n = (const float*)d_in[10];
  const float* bkqv_n = (const float*)d_in[11];
  const float* Woc    = (const float*)d_in[12];
  const float* boc    = (const float*)d_in[13];
  const float* Won    = (const float*)d_in[14];
  const float* bon    = (const float*)d_in[15];
  const float* Wkrel  = (const float*)d_in[16];
  const float* bkrel  = (const float*)d_in[17];
  const float* Wvrel  = (const float*)d_in[18];
  const float* bvrel  = (const float*)d_in[19];
  const float* p1     = (const float*)d_in[20];
  const float* p2     = (const float*)d_in[21];
  const float* lnwc   = (const float*)d_in[22];
  const float* lnbc   = (const float*)d_in[23];
  const float* lnwn   = (const float*)d_in[24];
  const float* lnbn   = (const float*)d_in[25];
  const float* wsel   = (const float*)d_in[26];

  const int F  = in_sizes[8] / 3;        // W_kqv_cell is (F,3)
  const int Nc = in_sizes[0] / F;
  const int Nn = in_sizes[1] / F;
  const int N  = Nc + Nn;
  const int E1 = in_sizes[2];
  const int E2 = in_sizes[4];
  const int POOL = out_size / F;
  int SORTN = 1; while (SORTN < N) SORTN <<= 1;

  // workspace layout
  char* ws = (char*)d_ws;
  size_t off = 0;
  auto alloc = [&](size_t bytes) -> void* {
    void* p = ws + off;
    off = (off + bytes + 255) & ~(size_t)255;
    return p;
  };
  float* qa    = (float*)alloc((size_t)N * 4);
  float* ka    = (float*)alloc((size_t)N * 4);
  float* va    = (float*)alloc((size_t)N * 4);
  float* amax  = (float*)alloc((size_t)N * 4);
  float* denom = (float*)alloc((size_t)N * 4);
  float* outn  = (float*)alloc((size_t)N * 4);
  float* yarr  = (float*)alloc((size_t)N * 4);
  float* score = (float*)alloc((size_t)N * 4);
  float* acc   = (float*)alloc(8 * 4);
  u64*   keys  = (u64*)  alloc((size_t)SORTN * 8);

  const int nb = (N + TPB - 1) / TPB;
  init_kernel<<<nb, TPB, 0, stream>>>(amax, denom, outn, acc, N);

  // kqv projections (WMMA): wave = 16-node tile, block = 8 tiles, B staged in LDS
  const size_t ldsB = (size_t)F * 16 * sizeof(float);
  {
    int tiles = (Nc + 15) / 16;
    int blk   = (tiles + (TPB / 32) - 1) / (TPB / 32);
    kqv_wmma_kernel<<<blk, TPB, ldsB, stream>>>(x_cell, Wkqv_c, bkqv_c, Wkrel, bkrel,
                                                Wvrel, bvrel, 0, Nc, F, 0, qa, ka, va);
  }
  {
    int tiles = (Nn + 15) / 16;
    int blk   = (tiles + (TPB / 32) - 1) / (TPB / 32);
    kqv_wmma_kernel<<<blk, TPB, ldsB, stream>>>(x_net, Wkqv_n, bkqv_n, Wkrel, bkrel,
                                                Wvrel, bvrel, 1, Nn, F, Nc, qa, ka, va);
  }

  // edge attention: max -> denom -> weighted sum (indices + q/k/v stay L2-resident)
  const int Et = E1 + E2;
  const int eb = (Et + TPB - 1) / TPB;
  edge_max_kernel<<<eb, TPB, 0, stream>>>(s1, d1, s2, d2, p1, p2, qa, ka, amax, E1, E2, Nc);
  edge_denom_kernel<<<eb, TPB, 0, stream>>>(s1, d1, s2, d2, p1, p2, qa, ka, amax, denom, E1, E2, Nc);
  edge_out_kernel<<<eb, TPB, 0, stream>>>(s1, d1, s2, d2, p1, p2, qa, ka, va, amax, denom, outn, E1, E2, Nc);

  // out-proj + per-type LN stats
  node_post_kernel<<<nb, TPB, 0, stream>>>(outn, Woc, boc, Won, bon, yarr, acc, N, Nc);

  // LN + gelu + tanh + sort keys (with padding)
  const int sb = SORTN / TPB;
  node_score_kernel<<<sb, TPB, 0, stream>>>(yarr, acc, lnwc, lnbc, lnwn, lnbn, wsel,
                                            score, keys, N, Nc, Nn, SORTN);

  // global descending bitonic sort (keys working set ~2MB -> lives in L2)
  for (int kk = 2; kk <= SORTN; kk <<= 1)
    for (int j = kk >> 1; j > 0; j >>= 1)
      bitonic_kernel<<<sb, TPB, 0, stream>>>(keys, j, kk);

  // out[p] = x_all[perm[p]] * vals[p]
  final_gather_kernel<<<POOL, TPB, 0, stream>>>(keys, score, x_cell, x_net,
                                                (float*)d_out, Nc, F);
}